// DGCNN_33354716020956
// MI455X (gfx1250) — compile-verified
//
#include <hip/hip_runtime.h>
#include <hip/hip_bf16.h>
#include <float.h>

#define NPTS 4096
#define NBAT 4
#define KNN  20

typedef float v2f __attribute__((ext_vector_type(2)));
typedef float v8f __attribute__((ext_vector_type(8)));

__device__ __forceinline__ float lrelu(float v) { return v > 0.f ? v : 0.2f * v; }

// monotonic float<->uint key for atomic max over signed floats
__device__ __forceinline__ unsigned fkey(float f) {
    unsigned u = __float_as_uint(f);
    return (u & 0x80000000u) ? ~u : (u | 0x80000000u);
}
__device__ __forceinline__ float funkey(unsigned k) {
    unsigned u = (k & 0x80000000u) ? (k ^ 0x80000000u) : ~k;
    return __uint_as_float(u);
}

// async global->LDS copy of one dword (gfx1250 VFLAT async op, ASYNCcnt)
__device__ __forceinline__ void async_g2l_b32(unsigned lds_addr,
                                              unsigned long long gaddr) {
    asm volatile("global_load_async_to_lds_b32 %0, %1, off"
                 :: "v"(lds_addr), "v"(gaddr) : "memory");
}
__device__ __forceinline__ void wait_async0() {
    asm volatile("s_wait_asynccnt 0" ::: "memory");
}

// ---------------------------------------------------------------------------
// padding helpers: x (B,N,3)->(B,N,4) with 0; ec1 w (64x6)->(64x8) Wa|0|Wb|0
// ---------------------------------------------------------------------------
__global__ void k_padx(const float* __restrict__ x, float* __restrict__ xp) {
    int i = blockIdx.x * blockDim.x + threadIdx.x;
    if (i >= NBAT * NPTS * 4) return;
    int c = i & 3, n = i >> 2;
    xp[i] = (c < 3) ? x[n * 3 + c] : 0.f;
}
__global__ void k_padw(const float* __restrict__ w, float* __restrict__ wp) {
    int i = blockIdx.x * blockDim.x + threadIdx.x;
    if (i >= 64 * 8) return;
    int c = i & 7, o = i >> 3;
    float v = 0.f;
    if (c < 3)              v = w[o * 6 + c];
    else if (c >= 4 && c < 7) v = w[o * 6 + 3 + (c - 4)];
    wp[i] = v;
}

// ---------------------------------------------------------------------------
// squared row norms: xx[b*N+n] = sum_c X[b,n,c]^2
// ---------------------------------------------------------------------------
__global__ void k_rownorm(const float* __restrict__ X, int lda, int C,
                          float* __restrict__ xx) {
    int i = blockIdx.x * blockDim.x + threadIdx.x;
    if (i >= NBAT * NPTS) return;
    const float* p = X + (size_t)i * lda;
    float s = 0.f;
    for (int c = 0; c < C; ++c) { float v = p[c]; s += v * v; }
    xx[i] = s;
}

// ---------------------------------------------------------------------------
// pairwise squared distances for one batch via f32 WMMA; K multiple of 4.
// one wave per 16x16 tile; 8 waves per block.
// ---------------------------------------------------------------------------
__global__ void k_dist(const float* __restrict__ X, int lda, int C,
                       const float* __restrict__ xx, float* __restrict__ dist,
                       int b) {
    int lane = threadIdx.x & 31, wave = threadIdx.x >> 5;
    int l16 = lane & 15, half = lane >> 4;
    int tile = blockIdx.x * 8 + wave;      // (N/16)^2 = 65536 tiles
    int it = tile >> 8;
    int jt = tile & 255;
    const float* Xb = X + (size_t)b * NPTS * lda;
    const float* ap = Xb + (size_t)(it * 16 + l16) * lda;
    const float* bp = Xb + (size_t)(jt * 16 + l16) * lda;
    v8f acc = {0.f, 0.f, 0.f, 0.f, 0.f, 0.f, 0.f, 0.f};
    for (int k = 0; k < C; k += 4) {
        int kk = k + 2 * half;
        v2f a  = *(const v2f*)(ap + kk);
        v2f bb = *(const v2f*)(bp + kk);
        acc = __builtin_amdgcn_wmma_f32_16x16x4_f32(false, a, false, bb,
                                                    (short)0, acc, false, false);
    }
    float xxj = xx[b * NPTS + jt * 16 + l16];
    #pragma unroll
    for (int v = 0; v < 8; ++v) {
        int i = it * 16 + v + 8 * half;
        float d = xx[b * NPTS + i] + xxj - 2.f * acc[v];
        dist[(size_t)i * NPTS + jt * 16 + l16] = d;
    }
}

// ---------------------------------------------------------------------------
// top-20 smallest per row (stable ties -> lower index, matches lax.top_k).
// one wave per row; per-lane sorted insert, then 20-round lane-min merge.
// ---------------------------------------------------------------------------
__global__ void k_topk(const float* __restrict__ dist, int* __restrict__ idx,
                       int b) {
    int lane = threadIdx.x & 31, wave = threadIdx.x >> 5;
    int row = blockIdx.x * 8 + wave;
    __shared__ float sd[8][32 * KNN];
    __shared__ int   si[8][32 * KNN];

    float dl[KNN]; int il[KNN];
    #pragma unroll
    for (int t = 0; t < KNN; ++t) { dl[t] = FLT_MAX; il[t] = 0x7fffffff; }

    const float* dr = dist + (size_t)row * NPTS;
    for (int j = lane; j < NPTS; j += 32) {
        float dd = dr[j]; int ii = j;
        if (dd <= dl[KNN - 1]) {
            #pragma unroll
            for (int t = 0; t < KNN; ++t) {
                bool sw = (dd < dl[t]) || (dd == dl[t] && ii < il[t]);
                float td = sw ? dl[t] : dd;  int ti = sw ? il[t] : ii;
                dl[t] = sw ? dd : dl[t];     il[t] = sw ? ii : il[t];
                dd = td; ii = ti;
            }
        }
    }
    #pragma unroll
    for (int t = 0; t < KNN; ++t) {
        sd[wave][lane * KNN + t] = dl[t];
        si[wave][lane * KNN + t] = il[t];
    }
    __syncthreads();

    int p = 0;
    int* outp = idx + ((size_t)b * NPTS + row) * KNN;
    for (int kk = 0; kk < KNN; ++kk) {
        float h  = (p < KNN) ? sd[wave][lane * KNN + p] : FLT_MAX;
        int   hi = (p < KNN) ? si[wave][lane * KNN + p] : 0x7fffffff;
        float mh = h; int mi = hi;
        #pragma unroll
        for (int off = 16; off >= 1; off >>= 1) {
            float oh = __shfl_xor(mh, off, 32);
            int   oi = __shfl_xor(mi, off, 32);
            if (oh < mh || (oh == mh && oi < mi)) { mh = oh; mi = oi; }
        }
        if (h == mh && hi == mi) p++;
        if (lane == 0) outp[kk] = mi;
    }
}

// ---------------------------------------------------------------------------
// generic f32 WMMA GEMM over per-point features, weight tile staged in LDS
// through async global->LDS loads.
//   mode 0: EdgeConv dual weights (cols [0,Cout): Wa-Wb, [Cout,2Cout): Wb)
//   mode 1: plain + optional bias/per-batch-bias/BN/LReLU, store col<collim
//   mode 2: like 1 but max over rows + atomic-max into gmkey[b*Nout+o]
// one wave per 16x16 tile; 8 waves/block share one weight tile (same nt).
// ---------------------------------------------------------------------------
__global__ void k_gemm(const float* __restrict__ A, int lda,
                       const float* __restrict__ W, int ldw, int wrows,
                       float* __restrict__ Out, int ldo,
                       int Kdim, int mode, int Cout,
                       const float* __restrict__ bias,
                       const float* __restrict__ pbias, int Nout,
                       const float* __restrict__ bng, const float* __restrict__ bnb,
                       const float* __restrict__ bnrm, const float* __restrict__ bnrv,
                       int act, int collim, unsigned* __restrict__ gmkey) {
    __shared__ alignas(16) float sW[16 * 514];   // 16 rows, stride KS+2 <= 514

    int lane = threadIdx.x & 31, wave = threadIdx.x >> 5;
    int l16 = lane & 15, half = lane >> 4;
    int mt = blockIdx.x * 8 + wave;
    int nt = blockIdx.y;
    int b  = blockIdx.z;

    bool ehalf1 = (mode == 0) && (nt * 16 >= Cout);   // second (Wb) half
    int KS      = (mode == 0 && !ehalf1) ? 2 * Kdim : Kdim;
    int rowbase = ehalf1 ? (nt * 16 - Cout) : nt * 16;
    int woff    = ehalf1 ? Kdim : 0;
    int stride  = KS + 2;

    // cooperative async stage of the 16 x KS weight tile into LDS
    for (int r = 0; r < 16; ++r) {
        int r2 = rowbase + r; if (r2 >= wrows) r2 = wrows - 1;
        const float* wp = W + (size_t)r2 * ldw + woff;
        float* lp = sW + r * stride;
        for (int c = threadIdx.x; c < KS; c += 256) {
            async_g2l_b32((unsigned)(size_t)(lp + c),
                          (unsigned long long)(size_t)(wp + c));
        }
    }
    wait_async0();
    __syncthreads();

    const float* ap = A + ((size_t)b * NPTS + mt * 16 + l16) * lda;
    const float* wr = sW + l16 * stride;
    v8f acc = {0.f, 0.f, 0.f, 0.f, 0.f, 0.f, 0.f, 0.f};
    for (int k = 0; k < Kdim; k += 4) {
        int kk = k + 2 * half;
        v2f a = *(const v2f*)(ap + kk);
        v2f bb;
        if (mode == 0 && !ehalf1) {
            v2f w0 = *(const v2f*)(wr + kk);
            v2f w1 = *(const v2f*)(wr + Kdim + kk);
            bb.x = w0.x - w1.x; bb.y = w0.y - w1.y;
        } else {
            bb = *(const v2f*)(wr + kk);
        }
        acc = __builtin_amdgcn_wmma_f32_16x16x4_f32(false, a, false, bb,
                                                    (short)0, acc, false, false);
    }

    int ocol = nt * 16 + l16;
    int pc = ocol < collim ? ocol : collim - 1;   // clamp param reads (seg3)
    float add = 0.f;
    if (bias)  add += bias[pc];
    if (pbias) add += pbias[(size_t)b * Nout + pc];
    float s = 1.f, t = 0.f;
    if (bng) {
        float sv = bng[pc] * rsqrtf(bnrv[pc] + 1e-5f);
        s = sv; t = bnb[pc] - bnrm[pc] * sv;
    }

    if (mode == 2) {
        float mx = -FLT_MAX;
        #pragma unroll
        for (int v = 0; v < 8; ++v) {
            float val = (acc[v] + add) * s + t;
            if (act) val = lrelu(val);
            mx = fmaxf(mx, val);
        }
        float o = __shfl_xor(mx, 16, 32);
        mx = fmaxf(mx, o);
        if (half == 0)
            atomicMax(gmkey + (size_t)b * Nout + ocol, fkey(mx));
    } else {
        #pragma unroll
        for (int v = 0; v < 8; ++v) {
            int i = mt * 16 + v + 8 * half;
            float val = acc[v] + add;
            if (bng) val = val * s + t;
            if (act) val = lrelu(val);
            if (ocol < collim)
                Out[((size_t)b * NPTS + i) * ldo + ocol] = val;
        }
    }
}

// ---------------------------------------------------------------------------
// EdgeConv epilogue: out[b,n,c] = max_j lrelu(bn(Y1[b,n,c] + Y2[b,idx[b,n,j],c]))
// ---------------------------------------------------------------------------
__global__ void k_edgemax(const float* __restrict__ Y, int Cout,
                          const int* __restrict__ idx,
                          const float* __restrict__ bng, const float* __restrict__ bnb,
                          const float* __restrict__ bnrm, const float* __restrict__ bnrv,
                          float* __restrict__ cat, int catoff) {
    int tid = threadIdx.x;
    int ppb = blockDim.x / Cout;
    int c = tid % Cout, pl = tid / Cout;
    int n = blockIdx.x * ppb + pl;
    int b = blockIdx.y;
    int ld2 = 2 * Cout;
    float y1 = Y[((size_t)b * NPTS + n) * ld2 + c];
    float sv = bng[c] * rsqrtf(bnrv[c] + 1e-5f);
    float tv = bnb[c] - bnrm[c] * sv;
    const int* ip = idx + ((size_t)b * NPTS + n) * KNN;
    float m = -FLT_MAX;
    #pragma unroll 4
    for (int j = 0; j < KNN; ++j) {
        int nb = ip[j];
        float y2 = Y[((size_t)b * NPTS + nb) * ld2 + Cout + c];
        m = fmaxf(m, lrelu((y1 + y2) * sv + tv));
    }
    cat[((size_t)b * NPTS + n) * 320 + catoff + c] = m;
}

// ---------------------------------------------------------------------------
// small helpers
// ---------------------------------------------------------------------------
__global__ void k_initkey(unsigned* gk) {
    int i = blockIdx.x * blockDim.x + threadIdx.x;
    if (i < NBAT * 1024) gk[i] = fkey(-FLT_MAX);
}
__global__ void k_decode(const unsigned* gk, float* gm) {
    int i = blockIdx.x * blockDim.x + threadIdx.x;
    if (i < NBAT * 1024) gm[i] = funkey(gk[i]);
}
// pb[b,o] = bias[o] + sum_c gm[b,c] * W[o, koff+c]
__global__ void k_pbias(const float* __restrict__ gm, const float* __restrict__ W,
                        int ldw, int koff, int kn, const float* __restrict__ bias,
                        float* __restrict__ pb, int Nout) {
    int o = blockIdx.x * blockDim.x + threadIdx.x;
    int b = blockIdx.y;
    if (o >= Nout) return;
    const float* wp = W + (size_t)o * ldw + koff;
    const float* g = gm + (size_t)b * 1024;
    float s = bias ? bias[o] : 0.f;
    for (int c = 0; c < kn; ++c) s += g[c] * wp[c];
    pb[(size_t)b * Nout + o] = s;
}

// ---------------------------------------------------------------------------
extern "C" void kernel_launch(void* const* d_in, const int* in_sizes, int n_in,
                              void* d_out, int out_size, void* d_ws, size_t ws_size,
                              hipStream_t stream) {
    (void)in_sizes; (void)n_in; (void)out_size; (void)ws_size;
    const float* x = (const float*)d_in[0];
    const float* ec_w[4]  = {(const float*)d_in[1],  (const float*)d_in[6],
                             (const float*)d_in[11], (const float*)d_in[16]};
    const float* ec_g[4]  = {(const float*)d_in[2],  (const float*)d_in[7],
                             (const float*)d_in[12], (const float*)d_in[17]};
    const float* ec_b[4]  = {(const float*)d_in[3],  (const float*)d_in[8],
                             (const float*)d_in[13], (const float*)d_in[18]};
    const float* ec_rm[4] = {(const float*)d_in[4],  (const float*)d_in[9],
                             (const float*)d_in[14], (const float*)d_in[19]};
    const float* ec_rv[4] = {(const float*)d_in[5],  (const float*)d_in[10],
                             (const float*)d_in[15], (const float*)d_in[20]};
    const float* glob_w  = (const float*)d_in[21];
    const float* glob_bi = (const float*)d_in[22];
    const float* glob_g  = (const float*)d_in[23];
    const float* glob_b  = (const float*)d_in[24];
    const float* glob_rm = (const float*)d_in[25];
    const float* glob_rv = (const float*)d_in[26];
    const float* s1_w  = (const float*)d_in[27];
    const float* s1_bi = (const float*)d_in[28];
    const float* s1_g  = (const float*)d_in[29];
    const float* s1_b  = (const float*)d_in[30];
    const float* s1_rm = (const float*)d_in[31];
    const float* s1_rv = (const float*)d_in[32];
    const float* s2_w  = (const float*)d_in[33];
    const float* s2_bi = (const float*)d_in[34];
    const float* s2_g  = (const float*)d_in[35];
    const float* s2_b  = (const float*)d_in[36];
    const float* s2_rm = (const float*)d_in[37];
    const float* s2_rv = (const float*)d_in[38];
    const float* s3_w  = (const float*)d_in[39];
    const float* s3_bi = (const float*)d_in[40];

    char* wsb = (char*)d_ws;
    size_t off = 0;
    auto alloc = [&](size_t bytes) {
        void* p = wsb + off;
        off += (bytes + 255) & ~(size_t)255;
        return p;
    };
    float*    xpad  = (float*)alloc((size_t)NBAT * NPTS * 4 * 4);
    float*    wpad  = (float*)alloc((size_t)64 * 8 * 4);
    float*    xx    = (float*)alloc((size_t)NBAT * NPTS * 4);
    float*    dist  = (float*)alloc((size_t)NPTS * NPTS * 4);     // per-batch reuse
    int*      idx   = (int*)  alloc((size_t)NBAT * NPTS * KNN * 4);
    float*    cat   = (float*)alloc((size_t)NBAT * NPTS * 320 * 4);
    float*    Y     = (float*)alloc((size_t)NBAT * NPTS * 256 * 4);
    unsigned* gmkey = (unsigned*)alloc((size_t)NBAT * 1024 * 4);
    float*    gm    = (float*)alloc((size_t)NBAT * 1024 * 4);
    float*    pb    = (float*)alloc((size_t)NBAT * 512 * 4);
    float*    h1    = (float*)alloc((size_t)NBAT * NPTS * 512 * 4);
    float*    h2    = (float*)alloc((size_t)NBAT * NPTS * 256 * 4);

    k_padx<<<(NBAT * NPTS * 4 + 255) / 256, 256, 0, stream>>>(x, xpad);
    k_padw<<<2, 256, 0, stream>>>(ec_w[0], wpad);

    const int Couts[4]  = {64, 64, 64, 128};
    const int Cins[4]   = {4, 64, 64, 64};    // ec1 padded to K=4
    const int catoff[4] = {0, 64, 128, 192};

    for (int L = 0; L < 4; ++L) {
        const float* X = (L == 0) ? xpad : (cat + catoff[L - 1]);
        int lda = (L == 0) ? 4 : 320;
        int C = Cins[L], Cout = Couts[L];
        const float* Wl = (L == 0) ? wpad : ec_w[L];
        int ldwl = 2 * C;                      // ec1: padded 8

        k_rownorm<<<(NBAT * NPTS + 255) / 256, 256, 0, stream>>>(X, lda, C, xx);
        for (int b = 0; b < NBAT; ++b) {
            k_dist<<<8192, 256, 0, stream>>>(X, lda, C, xx, dist, b);
            k_topk<<<NPTS / 8, 256, 0, stream>>>(dist, idx, b);
        }
        dim3 gg(NPTS / 16 / 8, (2 * Cout) / 16, NBAT);
        k_gemm<<<gg, 256, 0, stream>>>(X, lda, Wl, ldwl, Cout, Y, 2 * Cout,
                                       C, /*mode*/0, Cout,
                                       nullptr, nullptr, 2 * Cout,
                                       nullptr, nullptr, nullptr, nullptr,
                                       0, 2 * Cout, nullptr);
        dim3 ge(NPTS / (256 / Cout), NBAT);
        k_edgemax<<<ge, 256, 0, stream>>>(Y, Cout, idx,
                                          ec_g[L], ec_b[L], ec_rm[L], ec_rv[L],
                                          cat, catoff[L]);
    }

    // global feature: 320 -> 1024, bias+BN+LReLU, fused max over N
    k_initkey<<<(NBAT * 1024 + 255) / 256, 256, 0, stream>>>(gmkey);
    {
        dim3 gg(NPTS / 16 / 8, 1024 / 16, NBAT);
        k_gemm<<<gg, 256, 0, stream>>>(cat, 320, glob_w, 320, 1024, nullptr, 0,
                                       320, /*mode*/2, 0,
                                       glob_bi, nullptr, 1024,
                                       glob_g, glob_b, glob_rm, glob_rv,
                                       1, 1024, gmkey);
    }
    k_decode<<<(NBAT * 1024 + 255) / 256, 256, 0, stream>>>(gmkey, gm);

    // seg1: h1 = lrelu(bn(cat.W[:, :320]^T + (gm.W[:, 320:]^T + bias)))
    {
        dim3 gp((512 + 63) / 64, NBAT);
        k_pbias<<<gp, 64, 0, stream>>>(gm, s1_w, 1344, 320, 1024, s1_bi, pb, 512);
        dim3 gg(NPTS / 16 / 8, 512 / 16, NBAT);
        k_gemm<<<gg, 256, 0, stream>>>(cat, 320, s1_w, 1344, 512, h1, 512,
                                       320, /*mode*/1, 0,
                                       nullptr, pb, 512,
                                       s1_g, s1_b, s1_rm, s1_rv,
                                       1, 512, nullptr);
    }
    // seg2: 512 -> 256
    {
        dim3 gg(NPTS / 16 / 8, 256 / 16, NBAT);
        k_gemm<<<gg, 256, 0, stream>>>(h1, 512, s2_w, 512, 256, h2, 256,
                                       512, /*mode*/1, 0,
                                       s2_bi, nullptr, 256,
                                       s2_g, s2_b, s2_rm, s2_rv,
                                       1, 256, nullptr);
    }
    // seg3: 256 -> 13 (no BN, no act)
    {
        dim3 gg(NPTS / 16 / 8, 1, NBAT);
        k_gemm<<<gg, 256, 0, stream>>>(h2, 256, s3_w, 256, 13, (float*)d_out, 13,
                                       256, /*mode*/1, 0,
                                       s3_bi, nullptr, 16,
                                       nullptr, nullptr, nullptr, nullptr,
                                       0, 13, nullptr);
    }
}